// SelfGlobalPointer_10977936409142
// MI455X (gfx1250) — compile-verified
//
#include <hip/hip_runtime.h>
#include <hip/hip_bf16.h>

typedef __attribute__((ext_vector_type(2))) float v2f;
typedef __attribute__((ext_vector_type(8))) float v8f;

#define HEADS 12
#define HEAD_SIZE 64
#define HIDDEN 1024
#define BATCH 8
#define SEQLEN 512
#define NOUT (HEADS * 2 * HEAD_SIZE) /* 1536 */
#define INF_V 10000.0f

// ---------------------------------------------------------------------------
// Kernel 1: h = X @ W + b, fused bias + RoPE, scatter q/k to [B][H][S][64].
// One wave computes a 64x32 block of h (4 M-tiles x 2 N-tiles) with
// V_WMMA_F32_16X16X4_F32, amortizing the strided W loads 4x and X loads 2x.
// ---------------------------------------------------------------------------
__global__ __launch_bounds__(256) void qkv_rope_wmma(
    const float* __restrict__ X,      // [4096][1024]
    const float* __restrict__ W,      // [1024][1536]
    const float* __restrict__ bias,   // [1536]
    float* __restrict__ Qws,          // [B][H][S][64]
    float* __restrict__ Kws)          // [B][H][S][64]
{
    const int lane = threadIdx.x & 31;
    const int gw   = blockIdx.x * (blockDim.x >> 5) + (threadIdx.x >> 5);
    const int nBlkN = NOUT / 32;            // 48 column blocks (32 cols each)
    const int bm    = gw / nBlkN;           // 0..63  (64-row block)
    const int bn    = gw % nBlkN;           // 0..47

    const int half = lane >> 4;             // K-pair select / M-half select
    const int lmod = lane & 15;

    const float* aBase[4];
#pragma unroll
    for (int mt = 0; mt < 4; ++mt)
        aBase[mt] = X + (size_t)(bm * 64 + mt * 16 + lmod) * HIDDEN + half * 2;

    const float* bBase[2];
#pragma unroll
    for (int nt = 0; nt < 2; ++nt)
        bBase[nt] = W + (size_t)(half * 2) * NOUT + bn * 32 + nt * 16 + lmod;

    v8f c[4][2];
#pragma unroll
    for (int mt = 0; mt < 4; ++mt)
#pragma unroll
        for (int nt = 0; nt < 2; ++nt)
            c[mt][nt] = (v8f){};

#pragma unroll 2
    for (int k0 = 0; k0 < HIDDEN; k0 += 4) {
        v2f a[4];
#pragma unroll
        for (int mt = 0; mt < 4; ++mt)
            a[mt] = *(const v2f*)(aBase[mt] + k0);       // X[m][k0+2h..+1] b64
        v2f bv[2];
#pragma unroll
        for (int nt = 0; nt < 2; ++nt) {
            bv[nt].x = bBase[nt][(size_t)k0 * NOUT];     // W[k0+2h  ][n]
            bv[nt].y = bBase[nt][(size_t)(k0 + 1) * NOUT];
        }
#pragma unroll
        for (int mt = 0; mt < 4; ++mt)
#pragma unroll
            for (int nt = 0; nt < 2; ++nt)
                c[mt][nt] = __builtin_amdgcn_wmma_f32_16x16x4_f32(
                    false, a[mt], false, bv[nt], (short)0, c[mt][nt], false, false);
    }

    // --- bias + RoPE epilogue (lane owns column n; partner column = n^1 = lane^1)
#pragma unroll
    for (int nt = 0; nt < 2; ++nt) {
        const int   colB   = bn * 32 + nt * 16 + lmod;
        const float bval   = bias[colB];
        const int   head   = colB >> 7;     // n / 128
        const int   within = colB & 127;
        const int   isK    = within >> 6;   // 0 = q half, 1 = k half
        const int   d      = within & 63;
        const int   j      = d >> 1;
        const float inv_freq = __powf(10000.0f, -(2.0f * (float)j) / 64.0f);
        float* outBase = isK ? Kws : Qws;

#pragma unroll
        for (int mt = 0; mt < 4; ++mt) {
#pragma unroll
            for (int v = 0; v < 8; ++v) {
                const int   token   = bm * 64 + mt * 16 + v + 8 * half;
                const int   s       = token & (SEQLEN - 1);
                const int   bIdx    = token >> 9;        // token / 512
                const float val     = c[mt][nt][v] + bval;
                const float partner = __shfl_xor(val, 1, 32);
                float sn, cs;
                __sincosf((float)s * inv_freq, &sn, &cs);
                const float rot = (d & 1) ? (val * cs + partner * sn)
                                          : (val * cs - partner * sn);
                outBase[(((size_t)bIdx * HEADS + head) * SEQLEN + s) * HEAD_SIZE + d] = rot;
            }
        }
    }
}

// ---------------------------------------------------------------------------
// Kernel 2: logits[b][h] = Q(512x64) @ K^T(64x512), mask + scale, emit
// logits and boolean mask. One wave per 32x32 block (2x2 tiles of 16x16).
// ---------------------------------------------------------------------------
__global__ __launch_bounds__(256) void logits_wmma(
    const float* __restrict__ Qws,    // [B][H][S][64]
    const float* __restrict__ Kws,    // [B][H][S][64]
    const int*   __restrict__ mask,   // [B][S]
    float* __restrict__ outLogits,    // [B][H][S][S]
    float* __restrict__ outMask)      // [B][H][S][S]
{
    const int lane = threadIdx.x & 31;
    const int gw   = blockIdx.x * (blockDim.x >> 5) + (threadIdx.x >> 5);

    const int blk = gw & 255;               // 16x16 blocks of 32x32
    const int bh  = gw >> 8;                // b*HEADS + h
    const int b   = bh / HEADS;
    const int bn  = blk & 15;
    const int bm  = blk >> 4;

    const int half = lane >> 4;
    const int lmod = lane & 15;

    // A: Q[m][k], B: K^T[k][n] = K[n][k] — both contiguous along k -> b64 loads
    const float* qBase[2];
    const float* kBase[2];
#pragma unroll
    for (int t = 0; t < 2; ++t) {
        qBase[t] = Qws + ((size_t)bh * SEQLEN + bm * 32 + t * 16 + lmod) * HEAD_SIZE + half * 2;
        kBase[t] = Kws + ((size_t)bh * SEQLEN + bn * 32 + t * 16 + lmod) * HEAD_SIZE + half * 2;
    }

    v8f c[2][2];
#pragma unroll
    for (int mt = 0; mt < 2; ++mt)
#pragma unroll
        for (int nt = 0; nt < 2; ++nt)
            c[mt][nt] = (v8f){};

#pragma unroll
    for (int k0 = 0; k0 < HEAD_SIZE; k0 += 4) {
        v2f a[2], bv[2];
#pragma unroll
        for (int t = 0; t < 2; ++t) {
            a[t]  = *(const v2f*)(qBase[t] + k0);
            bv[t] = *(const v2f*)(kBase[t] + k0);
        }
#pragma unroll
        for (int mt = 0; mt < 2; ++mt)
#pragma unroll
            for (int nt = 0; nt < 2; ++nt)
                c[mt][nt] = __builtin_amdgcn_wmma_f32_16x16x4_f32(
                    false, a[mt], false, bv[nt], (short)0, c[mt][nt], false, false);
    }

    const size_t outBaseIdx = (size_t)bh * SEQLEN * SEQLEN;

#pragma unroll
    for (int nt = 0; nt < 2; ++nt) {
        const int   n  = bn * 32 + nt * 16 + lmod;      // key position (column)
        const float mN = (float)mask[b * SEQLEN + n];
#pragma unroll
        for (int mt = 0; mt < 2; ++mt) {
#pragma unroll
            for (int v = 0; v < 8; ++v) {
                const int   m    = bm * 32 + mt * 16 + v + 8 * half;  // query pos
                const float mM   = (float)mask[b * SEQLEN + m];
                const float pad  = 1.0f - mN * mM;
                const float tril = (n < m) ? 1.0f : 0.0f;  // tril(ones, k=-1)
                const float val  = (c[mt][nt][v] - pad * INF_V - tril * INF_V) * 0.125f;
                const size_t idx = outBaseIdx + (size_t)m * SEQLEN + n;
                outLogits[idx] = val;
                outMask[idx]   = (pad > 0.0f || tril > 0.0f) ? 1.0f : 0.0f;
            }
        }
    }
}

// ---------------------------------------------------------------------------
extern "C" void kernel_launch(void* const* d_in, const int* in_sizes, int n_in,
                              void* d_out, int out_size, void* d_ws, size_t ws_size,
                              hipStream_t stream) {
    const float* X    = (const float*)d_in[0];   // [8][512][1024]
    const float* W    = (const float*)d_in[1];   // [1024][1536]
    const float* bias = (const float*)d_in[2];   // [1536]
    const int*   mask = (const int*)d_in[3];     // [8][512]

    float* logits = (float*)d_out;
    float* omask  = logits + (size_t)BATCH * HEADS * SEQLEN * SEQLEN;

    float* Qws = (float*)d_ws;                                     // 12.6 MB
    float* Kws = Qws + (size_t)BATCH * HEADS * SEQLEN * HEAD_SIZE; // +12.6 MB

    // Kernel 1: (4096/64)*(1536/32) = 64*48 = 3072 waves, 8 waves/block
    qkv_rope_wmma<<<384, 256, 0, stream>>>(X, W, bias, Qws, Kws);

    // Kernel 2: 8*12*(16*16) = 24576 waves, 8 waves/block
    logits_wmma<<<3072, 256, 0, stream>>>(Qws, Kws, mask, logits, omask);
}